// PointPillarScatter_34806414967241
// MI455X (gfx1250) — compile-verified
//
#include <hip/hip_runtime.h>
#include <stdint.h>

// Problem constants (from the reference)
#define BB 4
#define CC 64
#define NN 100000
#define NY 496
#define NX 432
#define GRID_HW (NY * NX)          // 214272
#define INV_PILLAR 6.25f           // 1 / 0.16

typedef float v4f __attribute__((ext_vector_type(4)));

// ---- order-preserving float <-> uint key (so we can use integer atomicMin) ----
__device__ __forceinline__ unsigned f2key(float f) {
    unsigned b = __float_as_uint(f);
    return (b & 0x80000000u) ? ~b : (b | 0x80000000u);
}
__device__ __forceinline__ float key2f(unsigned k) {
    unsigned b = (k & 0x80000000u) ? (k ^ 0x80000000u) : ~k;
    return __uint_as_float(b);
}

// ---------------- kernel 1: init min slots ----------------
__global__ void pp_init_mins(unsigned* __restrict__ mins) {
    if (threadIdx.x < 2 * BB) mins[threadIdx.x] = 0xFFFFFFFFu;
}

// ---------------- kernel 2: per-batch min(x), min(y) ----------------
// points: (B*N, 4) float = [bidx, x, y, z]; read as float4 (16B coalesced).
__global__ __launch_bounds__(256) void pp_reduce_mins(
    const float4* __restrict__ pts, unsigned* __restrict__ mins)
{
    const int gp = blockIdx.x * 256 + threadIdx.x;
    unsigned kx = 0xFFFFFFFFu, ky = 0xFFFFFFFFu;
    if (gp < BB * NN) {
        float4 p = pts[gp];
        kx = f2key(p.y);   // x coordinate
        ky = f2key(p.z);   // y coordinate
    }
    // wave32 min-reduce
    #pragma unroll
    for (int off = 16; off > 0; off >>= 1) {
        unsigned ox = __shfl_xor(kx, off, 32);
        unsigned oy = __shfl_xor(ky, off, 32);
        kx = kx < ox ? kx : ox;
        ky = ky < oy ? ky : oy;
    }
    const int lane = threadIdx.x & 31;
    const int base = blockIdx.x * 256 + (threadIdx.x & ~31);  // wave-uniform
    if (lane == 0 && base < BB * NN) {
        const int b = base / NN;   // 32 | NN -> batch is wave-uniform
        atomicMin(&mins[2 * b + 0], kx);
        atomicMin(&mins[2 * b + 1], ky);
    }
}

// ---------------- kernel 3: zero the output grid (219 MB, NT stores) ----------------
__global__ __launch_bounds__(256) void pp_zero_out(v4f* __restrict__ out4, int n4) {
    const int i = blockIdx.x * 256 + threadIdx.x;
    if (i < n4) {
        v4f z = {0.f, 0.f, 0.f, 0.f};
        __builtin_nontemporal_store(z, &out4[i]);
    }
}

// ---------------- kernel 4: async-LDS scatter ----------------
// One wave handles 32 consecutive points. Grid = 3125 blocks x 128 threads
// = 400,000 = B*N exactly -> EXEC is all-ones everywhere (async ops want that).
// Since 32 | N, a wave never straddles a batch boundary.
__global__ __launch_bounds__(128) void pp_scatter(
    const float*   __restrict__ feat,   // (B, C, N)
    const float4*  __restrict__ pts,    // (B*N, 4)
    const unsigned* __restrict__ mins,  // 2*B keys
    float*         __restrict__ out)    // (B, C, NY, NX)
{
    __shared__ float tile[4][CC * 32];      // 8 KB per wave, 32 KB per block

    const int wave = threadIdx.x >> 5;
    const int lane = threadIdx.x & 31;
    const int base = blockIdx.x * 128 + wave * 32;   // wave-uniform
    const int b    = base / NN;                      // wave-uniform
    const int n0   = base - b * NN;                  // wave-uniform
    const int gp   = base + lane;

    // ---- stage the 64x32 feature tile into LDS with async b128 loads ----
    // Each lane moves 16B; 32 lanes cover 4 channel-rows (4 * 128B) per issue.
    {
        const int crow = lane >> 3;     // which of 4 rows this lane serves
        const int seg  = lane & 7;      // 16B segment within a 128B row
        const float* gsrc0 = feat + (size_t)b * CC * NN + n0 + seg * 4;
        const float* lsrc0 = &tile[wave][crow * 32 + seg * 4];
        #pragma unroll
        for (int c0 = 0; c0 < CC; c0 += 4) {
            unsigned lds_addr = (unsigned)(uintptr_t)(lsrc0 + c0 * 32);
            unsigned long long gaddr =
                (unsigned long long)(uintptr_t)(gsrc0 + (size_t)(c0 + crow) * NN);
            asm volatile("global_load_async_to_lds_b128 %0, %1, off"
                         :: "v"(lds_addr), "v"(gaddr) : "memory");
        }
    }

    // ---- overlap: compute pillar index while the tile is in flight ----
    float4 p = pts[gp];
    const float xmin = key2f(mins[2 * b + 0]);
    const float ymin = key2f(mins[2 * b + 1]);
    int xi = (int)floorf((p.y - xmin) * INV_PILLAR);
    int yi = (int)floorf((p.z - ymin) * INV_PILLAR);
    xi = xi < 0 ? 0 : (xi > NX - 1 ? NX - 1 : xi);
    yi = yi < 0 ? 0 : (yi > NY - 1 ? NY - 1 : yi);
    float* dst0 = out + (size_t)b * CC * GRID_HW + (size_t)yi * NX + xi;

    // loads must land in LDS before the async stores read it (per-wave tile,
    // no cross-wave sharing -> no barrier needed, just the counter)
    asm volatile("s_wait_asynccnt 0x0" ::: "memory");

    // ---- scattered writes: LDS -> global, one async b32 per channel ----
    #pragma unroll
    for (int c = 0; c < CC; ++c) {
        unsigned lds_addr = (unsigned)(uintptr_t)&tile[wave][c * 32 + lane];
        unsigned long long gaddr =
            (unsigned long long)(uintptr_t)(dst0 + (size_t)c * GRID_HW);
        asm volatile("global_store_async_from_lds_b32 %0, %1, off"
                     :: "v"(gaddr), "v"(lds_addr) : "memory");
    }
    asm volatile("s_wait_asynccnt 0x0" ::: "memory");
}

extern "C" void kernel_launch(void* const* d_in, const int* in_sizes, int n_in,
                              void* d_out, int out_size, void* d_ws, size_t ws_size,
                              hipStream_t stream) {
    const float*  feat = (const float*)d_in[0];          // (B, C, N)
    const float4* pts  = (const float4*)d_in[1];         // (B*N, 4)
    // d_in[2] (voxel_coords) is unused by the reference math.
    float*    out  = (float*)d_out;                      // (B, C, NY, NX)
    unsigned* mins = (unsigned*)d_ws;                    // 8 u32 slots

    // 1) seed the per-batch min slots
    pp_init_mins<<<1, 32, 0, stream>>>(mins);

    // 2) per-batch min(x), min(y)
    const int npts = BB * NN;                            // 400,000
    pp_reduce_mins<<<(npts + 255) / 256, 256, 0, stream>>>(pts, mins);

    // 3) zero the grid
    const int n4 = out_size / 4;                         // 13,713,408 float4
    pp_zero_out<<<(n4 + 255) / 256, 256, 0, stream>>>((v4f*)d_out, n4);

    // 4) async-LDS scatter: 3125 blocks x 128 threads == exactly B*N points
    pp_scatter<<<npts / 128, 128, 0, stream>>>(feat, pts, mins, out);
}